// ContinousConvolution_74929999446194
// MI455X (gfx1250) — compile-verified
//
#include <hip/hip_runtime.h>
#include <hip/hip_bf16.h>

#define NIN   20000
#define NOUT  20000
#define BB    64
#define EDGES 640000

typedef __attribute__((ext_vector_type(2))) float v2f;
typedef __attribute__((ext_vector_type(8))) float v8f;

// ---------------------------------------------------------------------------
// Kernel 1: xa_t[i*64 + b] = alpha[i] * x[b*NIN + i]   (transpose to src-major)
// Tile: 32 i-columns x 64 b-rows, LDS stride-33 padding -> conflict-free.
// ---------------------------------------------------------------------------
__global__ __launch_bounds__(256) void cc_prep_kernel(
    const float* __restrict__ x, const float* __restrict__ alpha,
    float* __restrict__ xa_t) {
  __shared__ float tile[64][33];
  const int tid = threadIdx.x;
  const int i0  = blockIdx.x * 32;
#pragma unroll
  for (int r = 0; r < 8; ++r) {
    int idx = r * 256 + tid;
    int b = idx >> 5, ii = idx & 31;          // lanes sweep ii -> coalesced x read
    int gi = i0 + ii;
    tile[b][ii] = alpha[gi] * x[b * NIN + gi];
  }
  __syncthreads();
#pragma unroll
  for (int r = 0; r < 8; ++r) {
    int idx = r * 256 + tid;
    int ii = idx >> 6, b = idx & 63;          // lanes sweep b -> coalesced write
    xa_t[(i0 + ii) * BB + b] = tile[b][ii];
  }
}

// ---------------------------------------------------------------------------
// Kernel 2: zero the [NOUT, 64] accumulator (float4 stores)
// ---------------------------------------------------------------------------
__global__ __launch_bounds__(256) void cc_zero_kernel(float4* __restrict__ p, int n4) {
  int i = blockIdx.x * blockDim.x + threadIdx.x;
  if (i < n4) p[i] = make_float4(0.f, 0.f, 0.f, 0.f);
}

// ---------------------------------------------------------------------------
// Kernel 3: edge scatter. One wave32 per 4 edges.
//   w_k = exp(-||p_in[src_k]-p_out[dst_k]||^2 / sigma[src_k]^2)
//   y_tmp[dst_k, b] += w_k * xa_t[src_k, b]   for b = 0..63
// Edge descriptors are wave-uniform: readfirstlane forces the edge id into an
// SGPR so index/position/sigma loads become SMEM and atomic bases become SADDR.
// WMMA: D(16x16,f32) = diag(w0..w3)[16x4] x B[4x16] per 16-wide b-chunk;
// D rows 0..3 are the scaled messages -> masked atomic scatter.
// ---------------------------------------------------------------------------
__global__ __launch_bounds__(256) void cc_edge_kernel(
    const long long* __restrict__ dst_idx, const long long* __restrict__ src_idx,
    const float* __restrict__ sigma, const float* __restrict__ ipos,
    const float* __restrict__ opos, const float* __restrict__ xa_t,
    float* __restrict__ y_tmp) {
  const int lane = threadIdx.x & 31;
  // uniform edge-group id -> SGPR (enables scalar loads for descriptors)
  const int eu = __builtin_amdgcn_readfirstlane(
      (int)(blockIdx.x * 8 + (threadIdx.x >> 5)) * 4);
  if (eu >= EDGES) return;

  int dst4[4], src4[4];
  float w4[4];
#pragma unroll
  for (int k = 0; k < 4; ++k) {
    const int d = (int)dst_idx[eu + k];       // uniform addr -> s_load
    const int s = (int)src_idx[eu + k];
    const float dx = ipos[s * 3 + 0] - opos[d * 3 + 0];
    const float dy = ipos[s * 3 + 1] - opos[d * 3 + 1];
    const float dz = ipos[s * 3 + 2] - opos[d * 3 + 2];
    const float sg = sigma[s];
    // fast divide: quotient feeds exp, 1-2 ulp is invisible after rounding
    w4[k] = __expf(-__fdividef(dx * dx + dy * dy + dz * dz, sg * sg));
    dst4[k] = d;
    src4[k] = s;
  }

#if __has_builtin(__builtin_amdgcn_wmma_f32_16x16x4_f32)
  // A (16x4 f32, 2 VGPRs): lanes 0-15 -> M=lane, v0:K=0, v1:K=1
  //                        lanes 16-31 -> M=lane-16, v0:K=2, v1:K=3
  // Diagonal: A[m,k] = w_k * (m == k)
  const int m = lane & 15;
  v2f a;
  if (lane < 16) {
    a.x = (m == 0) ? w4[0] : 0.f;
    a.y = (m == 1) ? w4[1] : 0.f;
  } else {
    a.x = (m == 2) ? w4[2] : 0.f;
    a.y = (m == 3) ? w4[3] : 0.f;
  }
  // B (4x16 f32, 2 VGPRs), row-striped across lanes (D-like half-lane split):
  // v0: K = 0 (lanes 0-15) / K = 2 (lanes 16-31); v1: K = 1 / K = 3
  const int n = lane & 15;
  const int k0 = (lane < 16) ? 0 : 2;
  const float* __restrict__ b0p = xa_t + (size_t)src4[k0] * BB + n;
  const float* __restrict__ b1p = xa_t + (size_t)src4[k0 + 1] * BB + n;
  // uniform atomic bases (SGPR) hoisted out of the chunk loop
  float* __restrict__ y0 = y_tmp + (size_t)dst4[0] * BB + n;
  float* __restrict__ y1 = y_tmp + (size_t)dst4[1] * BB + n;
  float* __restrict__ y2 = y_tmp + (size_t)dst4[2] * BB + n;
  float* __restrict__ y3 = y_tmp + (size_t)dst4[3] * BB + n;
#pragma unroll
  for (int c = 0; c < 4; ++c) {               // four 16-wide b-chunks cover B=64
    v2f bm;
    bm.x = b0p[c * 16];
    bm.y = b1p[c * 16];
    v8f cz = {};
    v8f dmat = __builtin_amdgcn_wmma_f32_16x16x4_f32(
        false, a, false, bm, (short)0, cz, false, false);
    // D layout: VGPR r = row r (lanes 0-15); rows 4..15 are zero by construction.
    if (lane < 16) {
      __hip_atomic_fetch_add(y0 + c * 16, dmat[0],
                             __ATOMIC_RELAXED, __HIP_MEMORY_SCOPE_AGENT);
      __hip_atomic_fetch_add(y1 + c * 16, dmat[1],
                             __ATOMIC_RELAXED, __HIP_MEMORY_SCOPE_AGENT);
      __hip_atomic_fetch_add(y2 + c * 16, dmat[2],
                             __ATOMIC_RELAXED, __HIP_MEMORY_SCOPE_AGENT);
      __hip_atomic_fetch_add(y3 + c * 16, dmat[3],
                             __ATOMIC_RELAXED, __HIP_MEMORY_SCOPE_AGENT);
    }
  }
#else
  // Fallback: lane handles b = 2*lane, 2*lane+1 (contiguous 64f per edge).
#pragma unroll
  for (int k = 0; k < 4; ++k) {
    const float* __restrict__ xp = xa_t + (size_t)src4[k] * BB;
    float* __restrict__ yp = y_tmp + (size_t)dst4[k] * BB;
    const float w = w4[k];
    const float v0 = xp[2 * lane];
    const float v1 = xp[2 * lane + 1];
    __hip_atomic_fetch_add(&yp[2 * lane], w * v0,
                           __ATOMIC_RELAXED, __HIP_MEMORY_SCOPE_AGENT);
    __hip_atomic_fetch_add(&yp[2 * lane + 1], w * v1,
                           __ATOMIC_RELAXED, __HIP_MEMORY_SCOPE_AGENT);
  }
#endif
}

// ---------------------------------------------------------------------------
// Kernel 4: d_out[b*NOUT + i] = y_tmp[i*64 + b]   (transpose back)
// ---------------------------------------------------------------------------
__global__ __launch_bounds__(256) void cc_outT_kernel(
    const float* __restrict__ y_tmp, float* __restrict__ out) {
  __shared__ float tile[64][33];
  const int tid = threadIdx.x;
  const int i0  = blockIdx.x * 32;
#pragma unroll
  for (int r = 0; r < 8; ++r) {
    int idx = r * 256 + tid;
    int ii = idx >> 6, b = idx & 63;          // lanes sweep b -> coalesced read
    tile[b][ii] = y_tmp[(size_t)(i0 + ii) * BB + b];
  }
  __syncthreads();
#pragma unroll
  for (int r = 0; r < 8; ++r) {
    int idx = r * 256 + tid;
    int b = idx >> 5, ii = idx & 31;          // lanes sweep ii -> coalesced write
    out[(size_t)b * NOUT + i0 + ii] = tile[b][ii];
  }
}

extern "C" void kernel_launch(void* const* d_in, const int* in_sizes, int n_in,
                              void* d_out, int out_size, void* d_ws, size_t ws_size,
                              hipStream_t stream) {
  const float*     x     = (const float*)d_in[0];      // [64, 20000]
  const float*     alpha = (const float*)d_in[1];      // [1, 20000]
  const float*     sigma = (const float*)d_in[2];      // [20000]
  const float*     ipos  = (const float*)d_in[3];      // [20000, 3]
  const float*     opos  = (const float*)d_in[4];      // [20000, 3]
  const long long* ei    = (const long long*)d_in[5];  // [2, 640000] int64
  float* out = (float*)d_out;

  float* xa_t  = (float*)d_ws;                         // [NIN, 64]  (5.12 MB)
  float* y_tmp = xa_t + (size_t)NIN * BB;              // [NOUT, 64] (5.12 MB)

  cc_prep_kernel<<<NIN / 32, 256, 0, stream>>>(x, alpha, xa_t);
  cc_zero_kernel<<<(NOUT * BB / 4 + 255) / 256, 256, 0, stream>>>(
      (float4*)y_tmp, NOUT * BB / 4);
  cc_edge_kernel<<<EDGES / 32, 256, 0, stream>>>(
      ei /*dst = row0*/, ei + EDGES /*src = row1*/, sigma, ipos, opos, xa_t, y_tmp);
  cc_outT_kernel<<<NOUT / 32, 256, 0, stream>>>(y_tmp, out);
}